// CTC_55293408969225
// MI455X (gfx1250) — compile-verified
//
#include <hip/hip_runtime.h>
#include <hip/hip_bf16.h>
#include <cstdint>

// Problem constants (fixed by the reference harness)
#define BB 16
#define TT 512
#define CC 4096
#define LL 128
#define SS 257            // 2*LL + 1
#define NEGV (-1e30f)

typedef __attribute__((ext_vector_type(2))) float v2f;
typedef __attribute__((ext_vector_type(8))) float v8f;

#if defined(__has_builtin)
#  if __has_builtin(__builtin_amdgcn_wmma_f32_16x16x4_f32)
#    define HAVE_WMMA_F32_4 1
#  endif
#endif

// ---------------------------------------------------------------------------
// Kernel 1: per-row (b,t) logsumexp denominator.
// One block = 256 threads (8 waves) handles 16 consecutive rows of C=4096.
// Pass 1: per-row max (coalesced float4 streams + LDS tree reduce).
// Pass 2: sum of exp via V_WMMA_F32_16X16X4_F32 with an all-ones B matrix:
//         D[m][n] += sum_k A[m][k]; exp runs on the TRANS pipe while the
//         matrix pipe accumulates. Two accumulators break the RAW chain.
//         Rows stay L2-resident between the two passes.
// ---------------------------------------------------------------------------
__global__ __launch_bounds__(256) void lse_denom_kernel(
    const float* __restrict__ scores, float* __restrict__ denom) {
  const int row0 = blockIdx.x * 16;          // global row = b*T + t
  const int tid  = threadIdx.x;
  const int lane = tid & 31;
  const int wave = tid >> 5;

  __shared__ float mpart[16][16];
  __shared__ float rowmax_s[16];
  __shared__ float spart[16][16];

  // ---- pass 1: row max ----
  {
    const int r   = tid >> 4;                // row within block (0..15)
    const int c16 = tid & 15;                // 16 threads per row
    const float4* rp = (const float4*)(scores + (size_t)(row0 + r) * CC);
    float m = -3.0e38f;
#pragma unroll 4
    for (int j = 0; j < 64; ++j) {           // 64 * 16 threads * 4 = 4096
      float4 v = rp[c16 + 16 * j];
      m = fmaxf(m, fmaxf(fmaxf(v.x, v.y), fmaxf(v.z, v.w)));
    }
    mpart[r][c16] = m;
  }
  __syncthreads();
  if (tid < 16) {
    float mm = mpart[tid][0];
#pragma unroll
    for (int j = 1; j < 16; ++j) mm = fmaxf(mm, mpart[tid][j]);
    rowmax_s[tid] = mm;
  }
  __syncthreads();

  // ---- pass 2: sum(exp(x - max)) ----
  const int mrow_i = lane & 15;              // A-matrix row handled by this lane
  const float mrow = rowmax_s[mrow_i];
  // wave w covers columns [w*512, (w+1)*512); lane holds K = 2*(lane>>4) + {0,1}
  const float* base = scores + (size_t)(row0 + mrow_i) * CC
                    + wave * 512 + 2 * (lane >> 4);

#if defined(HAVE_WMMA_F32_4)
  v8f acc0 = {};                             // two accumulators: break RAW chain
  v8f acc1 = {};
  v2f ones; ones.x = 1.0f; ones.y = 1.0f;    // B = ones(4x16)
#pragma unroll 4
  for (int it = 0; it < 128; it += 2) {
    float2 x0 = *(const float2*)(base + it * 4);
    float2 x1 = *(const float2*)(base + it * 4 + 4);
    v2f a0; a0.x = __expf(x0.x - mrow); a0.y = __expf(x0.y - mrow);
    v2f a1; a1.x = __expf(x1.x - mrow); a1.y = __expf(x1.y - mrow);
    acc0 = __builtin_amdgcn_wmma_f32_16x16x4_f32(
        false, a0, false, ones, (short)0, acc0, false, false);
    acc1 = __builtin_amdgcn_wmma_f32_16x16x4_f32(
        false, a1, false, ones, (short)0, acc1, false, false);
  }
  v8f acc = acc0 + acc1;                     // same D layout: row sums add
  __syncthreads();
  // D layout: VGPR j -> row j (lanes 0-15, N=lane) / row j+8 (lanes 16-31).
  // Column N=0 lives in lanes 0 and 16.
  if (lane == 0) {
#pragma unroll
    for (int j = 0; j < 8; ++j) spart[j][wave] = acc[j];
  }
  if (lane == 16) {
#pragma unroll
    for (int j = 0; j < 8; ++j) spart[j + 8][wave] = acc[j];
  }
  __syncthreads();
  if (tid < 16) {
    float s = 0.0f;
#pragma unroll
    for (int j = 0; j < 8; ++j) s += spart[tid][j];
    denom[row0 + tid] = rowmax_s[tid] + __logf(s);
  }
#else
  // VALU fallback: per-lane partial, unique (row, wave*2+hi) LDS slot.
  float part = 0.0f;
#pragma unroll 4
  for (int it = 0; it < 128; ++it) {
    float2 x = *(const float2*)(base + it * 4);
    part += __expf(x.x - mrow) + __expf(x.y - mrow);
  }
  __syncthreads();
  // lanes L and L+16 share mrow_i but differ in (lane>>4) -> unique slot.
  spart[mrow_i][wave * 2 + (lane >> 4)] = part;
  __syncthreads();
  if (tid < 16) {
    float s = 0.0f;
#pragma unroll
    for (int j = 0; j < 16; ++j) s += spart[tid][j];
    denom[row0 + tid] = rowmax_s[tid] + __logf(s);
  }
#endif
}

// ---------------------------------------------------------------------------
// Kernel 2: lp[b][t][s] = scores[b][t][ext(s)] - denom[b*T+t]
// ext(s) = labels[b][s/2] for odd s, else blank(0). Coalesced in s.
// ---------------------------------------------------------------------------
__global__ void gather_lp_kernel(const float* __restrict__ scores,
                                 const int* __restrict__ labels,
                                 const float* __restrict__ denom,
                                 float* __restrict__ lp) {
  const int idx = blockIdx.x * blockDim.x + threadIdx.x;
  const int total = BB * TT * SS;
  if (idx >= total) return;
  const int s  = idx % SS;
  const int bt = idx / SS;
  const int b  = bt / TT;
  const int ext = (s & 1) ? labels[b * LL + (s >> 1)] : 0;
  lp[idx] = scores[(size_t)bt * CC + ext] - denom[bt];
}

// ---------------------------------------------------------------------------
// Kernel 3: CTC alpha recursion. One block per batch, 288 threads (9 wave32),
// thread = extended symbol s. Alpha is double-buffered in LDS so each time
// step needs only ONE barrier (read cur / write cur^1 / barrier). lp is
// consumed through a double-buffered LDS tile of 8 time steps; the next
// tile's global loads are issued into registers before the current tile's
// steps are computed, hiding L2/HBM latency behind ~8 steps of compute.
// Early exit at feat_len (freeze semantics make later steps identity).
// ---------------------------------------------------------------------------
__global__ __launch_bounds__(288) void alpha_kernel(
    const float* __restrict__ lp, const int* __restrict__ labels,
    const int* __restrict__ feat_lens, const int* __restrict__ label_lens,
    float* __restrict__ loss_out) {
  const int b = blockIdx.x;
  const int s = threadIdx.x;
  const bool active = (s < SS);

  __shared__ float al[2][SS + 2];            // al[buf][s+2]; [0..1] guards=NEG
  __shared__ float lptile[2][8 * SS];        // 8 time steps per buffer

  const int flen = feat_lens[b];
  const int llen = label_lens[b];
  const float* lpb = lp + (size_t)b * TT * SS;

  bool skip = false;
  if (active && (s & 1) && s >= 3) {
    skip = (labels[b * LL + (s >> 1)] != labels[b * LL + (s >> 1) - 1]);
  }

  // init both alpha buffers to NEG, then alpha(t=0) into buffer 0
  if (s < SS + 2) { al[0][s] = NEGV; al[1][s] = NEGV; }
  __syncthreads();
  if (s == 0) al[0][2] = lpb[0];
  if (s == 1) al[0][3] = (llen > 0) ? lpb[1] : NEGV;
  __syncthreads();

  const int tmax = (flen < TT) ? flen : TT;  // steps t = 1 .. tmax-1
  const int maxflat = TT * SS - 1;

  float regs[8];
  // preload first chunk (t = 1..8) and stage to lptile[0]
#pragma unroll
  for (int k = 0; k < 8; ++k) {
    int f = s + 288 * k;
    int gi = 1 * SS + f; if (gi > maxflat) gi = maxflat;
    if (f < 8 * SS) regs[k] = lpb[gi];
  }
#pragma unroll
  for (int k = 0; k < 8; ++k) {
    int f = s + 288 * k;
    if (f < 8 * SS) lptile[0][f] = regs[k];
  }
  __syncthreads();

  int cur = 0;   // lp tile buffer
  int ca  = 0;   // alpha buffer
  for (int tc = 1; tc < tmax; tc += 8) {
    const int nexttc = tc + 8;
    const bool haveNext = (nexttc < tmax);
    if (haveNext) {
#pragma unroll
      for (int k = 0; k < 8; ++k) {          // issue next-chunk global loads now
        int f = s + 288 * k;
        int gi = nexttc * SS + f; if (gi > maxflat) gi = maxflat;
        if (f < 8 * SS) regs[k] = lpb[gi];
      }
    }
    const int steps = (tmax - tc < 8) ? (tmax - tc) : 8;
    for (int tt = 0; tt < steps; ++tt) {
      if (active) {
        const float lpv = lptile[cur][tt * SS + s];
        const float a  = al[ca][s + 2];
        const float a1 = al[ca][s + 1];
        const float a2 = skip ? al[ca][s] : NEGV;
        float mm = fmaxf(a, fmaxf(a1, a2));
        float newv = lpv + mm +
            __logf(__expf(a - mm) + __expf(a1 - mm) + __expf(a2 - mm));
        al[ca ^ 1][s + 2] = newv;            // write other buffer: no conflict
      }
      __syncthreads();                       // one barrier per time step
      ca ^= 1;
    }
    if (haveNext) {
#pragma unroll
      for (int k = 0; k < 8; ++k) {          // stage prefetched chunk
        int f = s + 288 * k;
        if (f < 8 * SS) lptile[cur ^ 1][f] = regs[k];
      }
    }
    __syncthreads();
    cur ^= 1;
  }

  if (s == 0) {
    const int end = 2 * llen;
    const float ae  = al[ca][end + 2];
    const float ae1 = (llen > 0) ? al[ca][end + 1] : NEGV;
    const float mm  = fmaxf(ae, ae1);
    const float ll  = mm + __logf(__expf(ae - mm) + __expf(ae1 - mm));
    float loss = -ll;
    if (loss > 0.5e30f) loss = 0.0f;         // zero_infinity
    const int dn = (llen > 1) ? llen : 1;
    loss_out[b] = loss / (float)dn;
  }
}

// ---------------------------------------------------------------------------
// Kernel 4: mean over batch -> scalar output.
// ---------------------------------------------------------------------------
__global__ void finalize_kernel(const float* __restrict__ loss_b,
                                float* __restrict__ out) {
  if (threadIdx.x == 0) {
    float sum = 0.0f;
#pragma unroll
    for (int i = 0; i < BB; ++i) sum += loss_b[i];
    out[0] = sum / (float)BB;
  }
}

extern "C" void kernel_launch(void* const* d_in, const int* in_sizes, int n_in,
                              void* d_out, int out_size, void* d_ws, size_t ws_size,
                              hipStream_t stream) {
  const float* scores     = (const float*)d_in[0];   // [B,T,C] f32
  const int*   labels     = (const int*)d_in[1];     // [B,L]
  const int*   feat_lens  = (const int*)d_in[2];     // [B]
  const int*   label_lens = (const int*)d_in[3];     // [B]
  float* ws    = (float*)d_ws;
  float* denom = ws;                                 // B*T = 8192 floats
  float* lp    = ws + BB * TT;                       // B*T*S floats (~8.4 MB)
  float* lossb = lp + (size_t)BB * TT * SS;          // B floats

  lse_denom_kernel<<<(BB * TT) / 16, 256, 0, stream>>>(scores, denom);
  const int total = BB * TT * SS;
  gather_lp_kernel<<<(total + 255) / 256, 256, 0, stream>>>(scores, labels, denom, lp);
  alpha_kernel<<<BB, 288, 0, stream>>>(lp, labels, feat_lens, label_lens, lossb);
  finalize_kernel<<<1, 32, 0, stream>>>(lossb, (float*)d_out);
}